// VolumeRenderer_26826365730885
// MI455X (gfx1250) — compile-verified
//
#include <hip/hip_runtime.h>
#include <cstdint>
#include <cstddef>

#define RPB   8      // rays per block: one ray per wave32
#define BLOCK 256    // 8 waves
#define NS    128    // coarse steps
#define TT    256    // merged samples per ray

typedef unsigned int u32x4 __attribute__((ext_vector_type(4)));
typedef int          i32x8 __attribute__((ext_vector_type(8)));
typedef int          i32x4 __attribute__((ext_vector_type(4)));
typedef float        f32x4 __attribute__((ext_vector_type(4)));

#if defined(__has_builtin)
#  if __has_builtin(__builtin_amdgcn_tensor_load_to_lds)
#    define HAVE_TDM 1
#  endif
#  if __has_builtin(__builtin_amdgcn_s_wait_tensorcnt)
#    define HAVE_TCNT 1
#  endif
#endif

__global__ __launch_bounds__(BLOCK) void nerf_resample_kernel(
    const float* __restrict__ rays_o,
    const float* __restrict__ rays_d,
    const float* __restrict__ weights,
    float* __restrict__ out,
    int n_rays)
{
  __shared__ float w_sh [RPB * NS];   // staged weight rows (TDM target)
  __shared__ float c_sh [RPB * NS];   // cdf, entries 0..126 used per ray
  __shared__ float nz_sh[RPB * NS];   // importance samples (sorted)
  __shared__ float za_sh[RPB * TT];   // merged z values

  const int lane = threadIdx.x & 31;
  const int wid  = threadIdx.x >> 5;
  const int rblk = blockIdx.x * RPB;
  const int r    = rblk + wid;
  const int rows = (n_rays - rblk < RPB) ? (n_rays - rblk) : RPB;

  // ---------- Phase 0: DMA this block's weight tile (rows x 128 f32) to LDS ----------
#if defined(HAVE_TDM)
  if (threadIdx.x == 0) {
    unsigned long long ga  = (unsigned long long)(weights + (size_t)rblk * NS);
    unsigned int       lds = (unsigned int)(unsigned long long)(&w_sh[0]);
    // D# group0: count=1 | lds_addr | global_addr[56:0] | type=2
    u32x4 g0 = { 1u, lds, (unsigned int)ga,
                 ((unsigned int)(ga >> 32) & 0x01FFFFFFu) | (2u << 30) };
    // D# group1: data_size=4B(2<<16) | tensor_dim0=128 | tensor_dim1=rows |
    //            tile_dim0=128 | tile_dim1=rows | tensor_dim0_stride=128
    i32x8 g1 = { (int)(2u << 16),
                 (int)(NS  << 16),
                 (int)(rows << 16),
                 (int)(NS  << 16),
                 rows,
                 NS,
                 0, 0 };
    i32x4 g2 = { 0, 0, 0, 0 };
    i32x4 g3 = { 0, 0, 0, 0 };
#if __clang_major__ >= 23
    __builtin_amdgcn_tensor_load_to_lds(g0, g1, g2, g3, (i32x8)0, 0);
#else
    __builtin_amdgcn_tensor_load_to_lds(g0, g1, g2, g3, 0);
#endif
#if defined(HAVE_TCNT)
    __builtin_amdgcn_s_wait_tensorcnt(0);
#endif
  }
#else
  { // fallback: cooperative coalesced staging
    const f32x4* src = reinterpret_cast<const f32x4*>(weights + (size_t)rblk * NS);
    f32x4*       dst = reinterpret_cast<f32x4*>(w_sh);
    const int n4 = rows * (NS / 4);
    for (int i = threadIdx.x; i < n4; i += BLOCK)
      dst[i] = __builtin_nontemporal_load(&src[i]);
  }
#endif
  // warm next block's tile while we compute (speculative, safe if OOB)
  __builtin_prefetch(weights + (size_t)(rblk + RPB) * NS, 0, 1);
  __syncthreads();

  const bool  active = (r < n_rays);
  const float step   = 2.0f / 127.0f;   // fars - nears == 2*BOUND always

  float* wrow  = &w_sh [wid * NS];
  float* crow  = &c_sh [wid * NS];
  float* nzrow = &nz_sh[wid * NS];
  float* zrow  = &za_sh[wid * TT];

  float ox=0.f, oy=0.f, oz=0.f, dx=0.f, dy=0.f, dz=0.f, nears=0.f;

  if (active) {
    ox = rays_o[3*r+0]; oy = rays_o[3*r+1]; oz = rays_o[3*r+2];
    dx = rays_d[3*r+0]; dy = rays_d[3*r+1]; dz = rays_d[3*r+2];
    nears = sqrtf(ox*ox + oy*oy + oz*oz) - 1.0f;

    // ---------- Phase 1: pdf + wave32 scan -> cdf[0..126] in LDS ----------
    float e[4]; float part = 0.0f;
#pragma unroll
    for (int t = 0; t < 4; ++t) {
      int k = lane * 4 + t;                              // e-index, 0..125 valid
      float v = (k < 126) ? (wrow[k + 1] + 1e-5f) : 0.0f; // weights[:,1:-1] + eps
      e[t] = v; part += v;
    }
    float incl = part;
#pragma unroll
    for (int off = 1; off < 32; off <<= 1) {
      float nb = __shfl_up(incl, off, 32);
      if (lane >= off) incl += nb;
    }
    float excl = incl - part;
    float invS = 1.0f / __shfl(incl, 31, 32);
    float run  = excl;
#pragma unroll
    for (int t = 0; t < 4; ++t) {
      int idx = lane * 4 + t + 1;
      run += e[t];
      if (idx <= 126) crow[idx] = run * invS;
    }
    if (lane == 0) crow[0] = 0.0f;
  }
  __syncthreads();

  if (active) {
    // ---------- Phase 2: deterministic inverse-CDF (128 samples, sorted u) ----------
#pragma unroll
    for (int t = 0; t < 4; ++t) {
      int   s = lane * 4 + t;
      float u = (s + 0.5f) * (1.0f / 128.0f);
      int l = 0, h = 126;                  // largest j with cdf[j] <= u
      while (h > l) { int m = (l + h + 1) >> 1; if (crow[m] <= u) l = m; else h = m - 1; }
      int   below = l, above = (l + 1 < 126) ? l + 1 : 126;
      float clo = crow[below], chi = crow[above];
      float den = chi - clo; if (den < 1e-5f) den = 1.0f;
      float tt  = (u - clo) / den;
      float blo = nears + step * (below + 0.5f);         // bins = z_vals_mid (affine)
      float bhi = nears + step * (above + 0.5f);
      nzrow[s] = blo + tt * (bhi - blo);
    }
  }
  __syncthreads();

  if (active) {
    // ---------- Phase 3: rank + scatter merge of two sorted 128-sequences ----------
    // new_z[s]: rank = #(z_vals <= x) + s   (z_vals affine, computed on the fly)
#pragma unroll
    for (int t = 0; t < 4; ++t) {
      int   s = lane * 4 + t;
      float x = nzrow[s];
      int l = -1, h = 127;
      while (h > l) { int m = (l + h + 1) >> 1; if (nears + step * m <= x) l = m; else h = m - 1; }
      zrow[(l + 1) + s] = x;
    }
    // z_vals[i]: rank = i + #(new_z < x)  (strict/non-strict split => permutation)
#pragma unroll
    for (int t = 0; t < 4; ++t) {
      int   i = lane * 4 + t;
      float x = nears + step * i;
      int l = -1, h = 127;
      while (h > l) { int m = (l + h + 1) >> 1; if (nzrow[m] < x) l = m; else h = m - 1; }
      zrow[i + (l + 1)] = x;
    }
  }
  __syncthreads();

  if (active) {
    // ---------- Phase 4: xyz = clamp(o + d*z), 96B/lane NT streaming stores ----------
    alignas(16) float buf[24];
#pragma unroll
    for (int q = 0; q < 8; ++q) {
      float z = zrow[lane * 8 + q];
      buf[q*3+0] = fminf(fmaxf(fmaf(dx, z, ox), -1.0f), 1.0f);
      buf[q*3+1] = fminf(fmaxf(fmaf(dy, z, oy), -1.0f), 1.0f);
      buf[q*3+2] = fminf(fmaxf(fmaf(dz, z, oz), -1.0f), 1.0f);
    }
    f32x4*       dst = reinterpret_cast<f32x4*>(out + (size_t)r * (TT * 3) + lane * 24);
    const f32x4* s4  = reinterpret_cast<const f32x4*>(buf);
#pragma unroll
    for (int q = 0; q < 6; ++q)
      __builtin_nontemporal_store(s4[q], &dst[q]);
  }
}

extern "C" void kernel_launch(void* const* d_in, const int* in_sizes, int n_in,
                              void* d_out, int out_size, void* d_ws, size_t ws_size,
                              hipStream_t stream) {
  (void)n_in; (void)out_size; (void)d_ws; (void)ws_size;
  const float* rays_o  = (const float*)d_in[0];
  const float* rays_d  = (const float*)d_in[1];
  const float* weights = (const float*)d_in[2];
  float*       out     = (float*)d_out;
  const int n_rays = in_sizes[0] / 3;
  const int blocks = (n_rays + RPB - 1) / RPB;
  hipLaunchKernelGGL(nerf_resample_kernel, dim3(blocks), dim3(BLOCK), 0, stream,
                     rays_o, rays_d, weights, out, n_rays);
}